// HSE_1915555414202
// MI455X (gfx1250) — compile-verified
//
#include <hip/hip_runtime.h>
#include <hip/hip_bf16.h>
#include <math.h>

// ---------------- problem constants (match reference) ----------------
#define GAT_C      128          // per-head channels
#define GAT_H      4            // heads
#define GAT_HC     512          // H*C
#define NEG_SLOPE  0.2f

typedef float v2f __attribute__((ext_vector_type(2)));
typedef float v8f __attribute__((ext_vector_type(8)));

#if defined(__has_builtin)
#if __has_builtin(__builtin_amdgcn_wmma_f32_16x16x4_f32)
#define GAT_HAVE_WMMA_F32 1
#endif
#endif

// ---------------------------------------------------------------------
// K1: h[M,512] = x[M,128] * W[128,512], full f32 via V_WMMA_F32_16X16X4_F32.
// One wave per 16x16 output tile, K marched in steps of 4 (32 WMMA / tile).
// A 16x4 layout: lanes 0-15 hold (K=k0,k0+1) of row M=lane; lanes 16-31 hold
// (K=k0+2,k0+3). B 4x16 layout: VGPR r, lanes 0-15 -> row k0+r, lanes 16-31
// -> row k0+2+r, column N=lane&15. C/D: VGPR r -> row r (+8 for upper lanes).
// ---------------------------------------------------------------------
__global__ __launch_bounds__(256) void gat_gemm_wmma(
    const float* __restrict__ x, const float* __restrict__ W,
    float* __restrict__ h, int n_nodes) {
  const int lane  = threadIdx.x & 31;
  const int wave  = threadIdx.x >> 5;            // 0..7
  const int mtile = blockIdx.x;
  const int ntile = blockIdx.y * 8 + wave;
  const int ntiles = GAT_HC / 16;                // 32
  if (ntile >= ntiles) return;
  const int m0 = mtile * 16;
  const int n0 = ntile * 16;
  const int half = lane >> 4;                    // 0 | 1
  const int l    = lane & 15;

#if GAT_HAVE_WMMA_F32
  int mload = m0 + l; if (mload >= n_nodes) mload = n_nodes - 1;
  const float* arow = x + (size_t)mload * GAT_C;
  const float* bcol = W + n0 + l;

  v8f acc = {};
#pragma unroll
  for (int k0 = 0; k0 < GAT_C; k0 += 4) {
    const int k = k0 + half * 2;
    v2f a, b;
    a.x = arow[k + 0];
    a.y = arow[k + 1];
    b.x = bcol[(size_t)(k + 0) * GAT_HC];
    b.y = bcol[(size_t)(k + 1) * GAT_HC];
    acc = __builtin_amdgcn_wmma_f32_16x16x4_f32(
        /*neg_a=*/false, a, /*neg_b=*/false, b,
        /*c_mod=*/(short)0, acc, /*reuse_a=*/false, /*reuse_b=*/false);
  }
  float* crow = h + n0 + l;
#pragma unroll
  for (int r = 0; r < 8; ++r) {
    const int m = m0 + r + 8 * half;
    if (m < n_nodes) crow[(size_t)m * GAT_HC] = acc[r];
  }
#else
  // scalar fallback (should not be taken on gfx1250 toolchains)
  float acc[8] = {0.f};
  for (int k = 0; k < GAT_C; ++k) {
    const float bkn = W[(size_t)k * GAT_HC + n0 + l];
#pragma unroll
    for (int r = 0; r < 8; ++r) {
      const int m = m0 + r + 8 * half;
      if (m < n_nodes) acc[r] = fmaf(x[(size_t)m * GAT_C + k], bkn, acc[r]);
    }
  }
#pragma unroll
  for (int r = 0; r < 8; ++r) {
    const int m = m0 + r + 8 * half;
    if (m < n_nodes) h[(size_t)m * GAT_HC + n0 + l] = acc[r];
  }
#endif
}

// ---------------------------------------------------------------------
// K2: a_src[n,h] = <h[n,h,:], att_src[h,:]> ; same for dst.
// Also (re)initialize emax = -inf, denom = 0 (graph-replay safe).
// ---------------------------------------------------------------------
__global__ __launch_bounds__(256) void gat_attn_dots(
    const float* __restrict__ h,
    const float* __restrict__ att_src, const float* __restrict__ att_dst,
    float* __restrict__ a_src, float* __restrict__ a_dst,
    float* __restrict__ emax, float* __restrict__ denom, int n_nodes) {
  const int t = blockIdx.x * blockDim.x + threadIdx.x;
  if (t >= n_nodes * GAT_H) return;
  const int n  = t >> 2;
  const int hd = t & 3;
  const float4* hv = (const float4*)(h + (size_t)n * GAT_HC + hd * GAT_C);
  const float4* as = (const float4*)(att_src + hd * GAT_C);
  const float4* ad = (const float4*)(att_dst + hd * GAT_C);
  float s1 = 0.f, s2 = 0.f;
#pragma unroll
  for (int i = 0; i < GAT_C / 4; ++i) {
    const float4 hh = hv[i], a1 = as[i], a2 = ad[i];
    s1 += hh.x * a1.x + hh.y * a1.y + hh.z * a1.z + hh.w * a1.w;
    s2 += hh.x * a2.x + hh.y * a2.y + hh.z * a2.z + hh.w * a2.w;
  }
  a_src[t] = s1;
  a_dst[t] = s2;
  emax[t]  = -INFINITY;
  denom[t] = 0.f;
}

// sign-split float atomic max (monotonic under IEEE-754 bit encodings)
__device__ __forceinline__ void atomicMaxF(float* addr, float value) {
  if (value >= 0.f)
    atomicMax((int*)addr, __float_as_int(value));
  else
    atomicMin((unsigned int*)addr, __float_as_uint(value));
}

__device__ __forceinline__ void gat_edge_ids(int e, const int* __restrict__ ei,
                                             int n_edges, int* s, int* d) {
  if (e < n_edges) { *s = ei[e]; *d = ei[n_edges + e]; }
  else             { *s = *d = e - n_edges; }         // self loop
}

__device__ __forceinline__ float gat_logit(const float* a_src, const float* a_dst,
                                           int s, int d, int hd) {
  const float v = a_src[s * GAT_H + hd] + a_dst[d * GAT_H + hd];
  return v > 0.f ? v : NEG_SLOPE * v;
}

// K3: segment max over destinations
__global__ __launch_bounds__(256) void gat_edge_max(
    const int* __restrict__ ei, const float* __restrict__ a_src,
    const float* __restrict__ a_dst, float* __restrict__ emax,
    int n_edges, int n_tot) {
  const int t = blockIdx.x * blockDim.x + threadIdx.x;
  if (t >= n_tot * GAT_H) return;
  const int e = t >> 2, hd = t & 3;
  int s, d; gat_edge_ids(e, ei, n_edges, &s, &d);
  atomicMaxF(&emax[d * GAT_H + hd], gat_logit(a_src, a_dst, s, d, hd));
}

// K4: ex = exp(e - emax[dst]); stash ex; denom[dst] += ex
__global__ __launch_bounds__(256) void gat_edge_exp(
    const int* __restrict__ ei, const float* __restrict__ a_src,
    const float* __restrict__ a_dst, const float* __restrict__ emax,
    float* __restrict__ denom, float* __restrict__ exbuf,
    int n_edges, int n_tot) {
  const int t = blockIdx.x * blockDim.x + threadIdx.x;
  if (t >= n_tot * GAT_H) return;
  const int e = t >> 2, hd = t & 3;
  int s, d; gat_edge_ids(e, ei, n_edges, &s, &d);
  const float ex = __expf(gat_logit(a_src, a_dst, s, d, hd) - emax[d * GAT_H + hd]);
  exbuf[t] = ex;
  atomicAdd(&denom[d * GAT_H + hd], ex);
}

// K5: alpha = ex / (denom[dst] + 1e-16)
__global__ __launch_bounds__(256) void gat_edge_norm(
    const int* __restrict__ ei, const float* __restrict__ denom,
    float* __restrict__ exbuf, int n_edges, int n_tot) {
  const int t = blockIdx.x * blockDim.x + threadIdx.x;
  if (t >= n_tot * GAT_H) return;
  const int e = t >> 2, hd = t & 3;
  int s, d; gat_edge_ids(e, ei, n_edges, &s, &d);
  exbuf[t] = exbuf[t] / (denom[d * GAT_H + hd] + 1e-16f);
}

// K6: out = bias (broadcast init each call)
__global__ __launch_bounds__(256) void gat_bias_init(
    const float* __restrict__ bias, float* __restrict__ out, int n_nodes) {
  const int t = blockIdx.x * blockDim.x + threadIdx.x;
  if (t < n_nodes * GAT_HC) out[t] = bias[t & (GAT_HC - 1)];
}

// K7: out[dst] += alpha * h[src] — 128 threads per edge, float4 per thread.
// h and out are L2-resident (20.5 MB each vs 192 MB L2) => on-chip scatter.
__global__ __launch_bounds__(256) void gat_aggregate(
    const int* __restrict__ ei, const float* __restrict__ h,
    const float* __restrict__ alpha, float* __restrict__ out,
    int n_edges, int n_tot) {
  const long long tid = (long long)blockIdx.x * blockDim.x + threadIdx.x;
  if (tid >= (long long)n_tot * 128) return;
  const int e  = (int)(tid >> 7);
  const int t  = (int)(tid & 127);
  const int hd = t >> 5;
  const int c4 = (t & 31) << 2;
  int s, d; gat_edge_ids(e, ei, n_edges, &s, &d);
  const float al = alpha[e * GAT_H + hd];
  const float4 hv = *(const float4*)(h + (size_t)s * GAT_HC + hd * GAT_C + c4);
  float* o = out + (size_t)d * GAT_HC + hd * GAT_C + c4;
  atomicAdd(o + 0, al * hv.x);
  atomicAdd(o + 1, al * hv.y);
  atomicAdd(o + 2, al * hv.z);
  atomicAdd(o + 3, al * hv.w);
}

// ---------------------------------------------------------------------
extern "C" void kernel_launch(void* const* d_in, const int* in_sizes, int n_in,
                              void* d_out, int out_size, void* d_ws, size_t ws_size,
                              hipStream_t stream) {
  const float* x       = (const float*)d_in[0];   // [N,128]
  const float* W       = (const float*)d_in[1];   // [128,512]
  const float* att_src = (const float*)d_in[2];   // [4,128]
  const float* att_dst = (const float*)d_in[3];   // [4,128]
  const float* bias    = (const float*)d_in[4];   // [512]
  const int*   ei      = (const int*)d_in[5];     // [2,E] (src row, dst row)

  const int n_nodes = in_sizes[0] / GAT_C;        // 10000
  const int n_edges = in_sizes[5] / 2;            // 320000
  const int n_tot   = n_edges + n_nodes;          // edges + self loops

  // workspace carve-up
  char* ws = (char*)d_ws;
  float* h      = (float*)ws;                                   // N*512 f32
  float* a_src  = (float*)(ws + (size_t)n_nodes * GAT_HC * 4);
  float* a_dst  = a_src + (size_t)n_nodes * GAT_H;
  float* emax   = a_dst + (size_t)n_nodes * GAT_H;
  float* denom  = emax  + (size_t)n_nodes * GAT_H;
  float* exbuf  = denom + (size_t)n_nodes * GAT_H;              // n_tot*4 f32
  float* out    = (float*)d_out;

  // K1: WMMA f32 GEMM
  {
    dim3 grid((n_nodes + 15) / 16, (GAT_HC / 16 + 7) / 8);
    gat_gemm_wmma<<<grid, 256, 0, stream>>>(x, W, h, n_nodes);
  }
  // K2: attention dots + state init
  {
    const int nt = n_nodes * GAT_H;
    gat_attn_dots<<<(nt + 255) / 256, 256, 0, stream>>>(
        h, att_src, att_dst, a_src, a_dst, emax, denom, n_nodes);
  }
  const int eht = n_tot * GAT_H;
  // K3: segment max
  gat_edge_max<<<(eht + 255) / 256, 256, 0, stream>>>(
      ei, a_src, a_dst, emax, n_edges, n_tot);
  // K4: exp + denom
  gat_edge_exp<<<(eht + 255) / 256, 256, 0, stream>>>(
      ei, a_src, a_dst, emax, denom, exbuf, n_edges, n_tot);
  // K5: normalize -> alpha
  gat_edge_norm<<<(eht + 255) / 256, 256, 0, stream>>>(
      ei, denom, exbuf, n_edges, n_tot);
  // K6: out = bias
  {
    const int nt = n_nodes * GAT_HC;
    gat_bias_init<<<(nt + 255) / 256, 256, 0, stream>>>(bias, out, n_nodes);
  }
  // K7: weighted scatter aggregation
  {
    const long long nt = (long long)n_tot * 128;
    const int blocks = (int)((nt + 255) / 256);
    gat_aggregate<<<blocks, 256, 0, stream>>>(ei, h, exbuf, out, n_edges, n_tot);
  }
}